// GaussianDynamics_RecurrentCell_64192581206140
// MI455X (gfx1250) — compile-verified
//
#include <hip/hip_runtime.h>
#include <hip/hip_bf16.h>
#include <stdint.h>

// GaussianDynamics recurrent scan, MI455X (gfx1250).
//
// Memory-bound (0.9 flop/byte vs ~2.6 machine balance): optimize for the
// 23.3 TB/s HBM path. One thread owns one batch element's serial scan.
// Triple-buffered async DMA (GLOBAL_LOAD_ASYNC_TO_LDS_B128, ASYNCcnt) keeps
// ~2 chunks (~15.7 MB chip-wide) of HBM requests perpetually in flight via a
// partial s_wait_asynccnt threshold (async loads complete in issue order).
// WMMA is intentionally NOT used: 2x2 matrices would occupy 1.5% of a 16x16
// WMMA tile; the CDNA5 feature this workload rewards is the async DMA path.

#define BLOCK 128          // threads per workgroup (4 waves, wave32)
#define KSTEP 20           // time steps per chunk (even; 1000 % 20 == 0)
#define NBUF  3            // triple buffering
#define NISSUE (KSTEP + KSTEP / 2)   // async loads issued per wave per chunk = 30

static_assert((KSTEP % 2) == 0, "dy packing needs even KSTEP");
static_assert(NISSUE == 30, "update s_wait_asynccnt immediates (0x1e) if changed");

// LDS layout per buffer: [ xtile: KSTEP*BLOCK float4 ][ dtile: (KSTEP/2)*BLOCK float4 ]
// xtile[k*BLOCK + b_local] = xicov(b,t0+k) 2x2 row-major
// dtile[m*BLOCK + b_local] = {dy(b,t0+2m), dy(b,t0+2m+1)}
#define XT_BYTES   (KSTEP * BLOCK * 16u)
#define DT_BYTES   ((KSTEP / 2) * BLOCK * 16u)
#define BUF_BYTES  (XT_BYTES + DT_BYTES)
#define LDS_BYTES  ((unsigned)NBUF * BUF_BYTES)   // 184320 B (<= 320 KB / WGP)

__global__ __launch_bounds__(BLOCK)
void gd_scan_kernel(const float* __restrict__ xicovs,   // [B,T,2,2]
                    const float* __restrict__ dyp,      // [B,T,2]
                    const float* __restrict__ cA,       // [2,2]
                    const float* __restrict__ Cm,       // [2,2]
                    const float* __restrict__ x0p,      // [B,2]
                    const float* __restrict__ dtp,      // [1]
                    float* __restrict__ out,            // [B,T,2] ++ [B,2]
                    int B, int T)
{
    extern __shared__ __align__(16) char smem[];
    const int tid = threadIdx.x;
    const int b0  = blockIdx.x * BLOCK;
    const int b   = b0 + tid;
    const bool valid = (b < B);

    // Uniform per-step operators (hoisted, algebraically equal to reference):
    //   out   = (C*dt) @ x
    //   x_new = (I + A*dt) @ x + xic @ (dy - out)
    const float dtv  = dtp[0];
    const float Cd00 = Cm[0] * dtv, Cd01 = Cm[1] * dtv;
    const float Cd10 = Cm[2] * dtv, Cd11 = Cm[3] * dtv;
    const float P00  = 1.0f + cA[0] * dtv, P01 = cA[1] * dtv;
    const float P10  = cA[2] * dtv,        P11 = 1.0f + cA[3] * dtv;

    float xr0 = 0.0f, xr1 = 0.0f;
    if (valid) { xr0 = x0p[2 * b]; xr1 = x0p[2 * b + 1]; }

    // Low 32 bits of a generic LDS pointer == wave-relative LDS byte offset.
    const unsigned sbase = (unsigned)(uintptr_t)(void*)smem;

    // Workgroup-uniform global bases (land in SGPR pairs for saddr form).
    const uint64_t xgbase = (uint64_t)(uintptr_t)xicovs + (uint64_t)b0 * (uint64_t)T * 16u;
    const uint64_t dgbase = (uint64_t)(uintptr_t)dyp    + (uint64_t)b0 * (uint64_t)T * 8u;

    const int nc = T / KSTEP;

    // Issue one chunk's async DMA into LDS buffer p. Exactly NISSUE async-load
    // instructions per wave, unconditionally (required for the suffix-counting
    // s_wait_asynccnt threshold). Out-of-range batch rows read from offset 0
    // into their own (never consumed) LDS slot.
    auto issue = [&](int c, int p) {
        const int t0 = c * KSTEP;
        const unsigned xl = sbase + (unsigned)p * BUF_BYTES;
        const unsigned dl = xl + XT_BYTES;
#pragma unroll
        for (int i = 0; i < KSTEP; ++i) {                // xicov: 16B per (b,t)
            const int s  = i * BLOCK + tid;
            const int bl = s / KSTEP;
            const int k  = s - bl * KSTEP;
            const bool ok = (b0 + bl < B);
            const unsigned voff = ok ? (unsigned)((bl * T + t0 + k) * 16) : 0u;
            const unsigned lds  = xl + (unsigned)((k * BLOCK + bl) * 16);
            asm volatile("global_load_async_to_lds_b128 %0, %1, %2 th:TH_LOAD_NT"
                         :: "v"(lds), "v"(voff), "s"(xgbase) : "memory");
        }
#pragma unroll
        for (int i = 0; i < KSTEP / 2; ++i) {            // dy: 16B covers 2 steps
            const int s  = i * BLOCK + tid;
            const int bl = s / (KSTEP / 2);
            const int m  = s - bl * (KSTEP / 2);
            const bool ok = (b0 + bl < B);
            const unsigned voff = ok ? (unsigned)(bl * T * 8 + (t0 + 2 * m) * 8) : 0u;
            const unsigned lds  = dl + (unsigned)((m * BLOCK + bl) * 16);
            asm volatile("global_load_async_to_lds_b128 %0, %1, %2 th:TH_LOAD_NT"
                         :: "v"(lds), "v"(voff), "s"(dgbase) : "memory");
        }
    };

    // Prologue: prime two chunks, then wait for chunk 0 only.
    if (nc > 0) issue(0, 0);
    if (nc > 1) {
        issue(1, 1);
        asm volatile("s_wait_asynccnt 0x1e" ::: "memory");  // <=30 pending => chunk 0 done
    } else {
        asm volatile("s_wait_asynccnt 0x0" ::: "memory");
    }
    __syncthreads();

    for (int c = 0; c < nc; ++c) {
        const int p = c % NBUF;
        const bool deep = (c + 2 < nc);
        if (deep) issue(c + 2, (c + 2) % NBUF);   // keep 2 chunks of DMA in flight

        if (valid) {
            const int t0 = c * KSTEP;
            const char* xl = smem + (size_t)p * BUF_BYTES;
            const char* dl = xl + XT_BYTES;
            unsigned long long* op =
                (unsigned long long*)(out + ((size_t)b * T + t0) * 2);
#pragma unroll
            for (int k = 0; k < KSTEP; ++k) {
                const float4 a = *(const float4*)(xl + (size_t)(k * BLOCK + tid) * 16);
                const float* dv = (const float*)(dl + (size_t)((k >> 1) * BLOCK + tid) * 16)
                                  + (k & 1) * 2;
                const float d0 = dv[0], d1 = dv[1];

                const float o0 = fmaf(Cd00, xr0, Cd01 * xr1);
                const float o1 = fmaf(Cd10, xr0, Cd11 * xr1);
                const float u0 = d0 - o0, u1 = d1 - o1;
                const float n0 = fmaf(P00, xr0, fmaf(P01, xr1, fmaf(a.x, u0, a.y * u1)));
                const float n1 = fmaf(P10, xr0, fmaf(P11, xr1, fmaf(a.z, u0, a.w * u1)));

                union { float f[2]; unsigned long long u; } pk;
                pk.f[0] = o0; pk.f[1] = o1;
                __builtin_nontemporal_store(pk.u, op + k);  // single-use stream
                xr0 = n0; xr1 = n1;
            }
        }

        // In-order async completion: <=NISSUE pending => chunk c+1 fully landed,
        // while chunk c+2's DMA keeps streaming through the wait.
        if (deep) asm volatile("s_wait_asynccnt 0x1e" ::: "memory");
        else      asm volatile("s_wait_asynccnt 0x0"  ::: "memory");
        __syncthreads();                                    // all waves done with buf p
    }

    // Tail (T % KSTEP != 0): direct streaming loads, same math.
    if (valid) {
        for (int t = nc * KSTEP; t < T; ++t) {
            const float4 a = ((const float4*)xicovs)[(size_t)b * T + t];
            const float2 d = ((const float2*)dyp)[(size_t)b * T + t];
            const float o0 = fmaf(Cd00, xr0, Cd01 * xr1);
            const float o1 = fmaf(Cd10, xr0, Cd11 * xr1);
            const float u0 = d.x - o0, u1 = d.y - o1;
            const float n0 = fmaf(P00, xr0, fmaf(P01, xr1, fmaf(a.x, u0, a.y * u1)));
            const float n1 = fmaf(P10, xr0, fmaf(P11, xr1, fmaf(a.z, u0, a.w * u1)));
            union { float f[2]; unsigned long long u; } pk;
            pk.f[0] = o0; pk.f[1] = o1;
            __builtin_nontemporal_store(pk.u,
                (unsigned long long*)out + (size_t)b * T + t);
            xr0 = n0; xr1 = n1;
        }
        // final state, appended after outputs
        out[(size_t)B * T * 2 + 2 * b]     = xr0;
        out[(size_t)B * T * 2 + 2 * b + 1] = xr1;
    }
}

extern "C" void kernel_launch(void* const* d_in, const int* in_sizes, int n_in,
                              void* d_out, int out_size, void* d_ws, size_t ws_size,
                              hipStream_t stream) {
    (void)n_in; (void)out_size; (void)d_ws; (void)ws_size;
    const float* xicovs = (const float*)d_in[0];
    const float* dy     = (const float*)d_in[1];
    const float* cA     = (const float*)d_in[2];
    const float* Cm     = (const float*)d_in[3];
    const float* x0     = (const float*)d_in[4];
    const float* dt     = (const float*)d_in[5];

    const int B = in_sizes[4] / 2;                 // x0 is [B,2]
    const int T = (B > 0) ? in_sizes[1] / (2 * B)  // dy is [B,T,2]
                          : 0;

    const int grid = (B + BLOCK - 1) / BLOCK;
    gd_scan_kernel<<<grid, BLOCK, LDS_BYTES, stream>>>(
        xicovs, dy, cA, Cm, x0, dt, (float*)d_out, B, T);
}